// GraphConvProlongation_51187420234090
// MI455X (gfx1250) — compile-verified
//
#include <hip/hip_runtime.h>
#include <math.h>

// Problem constants (match reference).
#define BB   4
#define CIN  16
#define COUT 16
#define NN   16384
#define EE   131072

typedef __attribute__((ext_vector_type(2))) float v2f;
typedef __attribute__((ext_vector_type(8))) float v8f;

// ---------------------------------------------------------------------------
// K0: zero the scatter accumulators (agg[N][16] + cnt[N]) in workspace.
// ---------------------------------------------------------------------------
__global__ void zero_ws_kernel(float* __restrict__ ws, int n) {
  int i = blockIdx.x * blockDim.x + threadIdx.x;
  int stride = gridDim.x * blockDim.x;
  for (; i < n; i += stride) ws[i] = 0.0f;
}

// ---------------------------------------------------------------------------
// K0b: transpose batch-0 features u[0][c][n] -> xT[n][c] (64B-aligned rows)
// so the per-edge gather is 1 cacheline instead of 16 scattered touches.
// Reads are fully coalesced (consecutive n across lanes); writes merge into
// 4x global_store_b128 per thread.
// ---------------------------------------------------------------------------
__global__ __launch_bounds__(256) void transpose_x0_kernel(
    const float* __restrict__ u, float* __restrict__ xT) {
  const int n = blockIdx.x * 256 + threadIdx.x;
#pragma unroll
  for (int c = 0; c < 16; ++c) xT[n * 16 + c] = u[c * NN + n];
}

// ---------------------------------------------------------------------------
// K1: per-edge kernel. Each lane owns one edge for the gather + 36->8 MLP;
// each wave then runs two 16-edge WMMA subtiles for h@W2 fused with the
// per-edge matvec, and scatters msg with coalesced f32 atomics.
// ---------------------------------------------------------------------------
__global__ __launch_bounds__(256) void edge_kernel(
    const float* __restrict__ xT,    // [N][16]  batch-0 features, row-major
    const float* __restrict__ grid,  // [B][N][2], only batch 0 used here
    const int*   __restrict__ ei,    // [2][E]
    const float* __restrict__ kw1,   // [36][8]
    const float* __restrict__ kb1,   // [8]
    const float* __restrict__ kw2,   // [8][256]
    const float* __restrict__ kb2,   // [256]
    float* __restrict__ agg,         // [N][16]
    float* __restrict__ cnt) {       // [N]
  __shared__ float kw1s[36 * 8];
  __shared__ float kb1s[8];
  __shared__ float kw2s[8 * 256];
  __shared__ float kb2s[256];
  __shared__ float xs[8][16 * 32];   // per-wave x[src] tiles, TRANSPOSED [c][edge]
  __shared__ float hs[8][32 * 8];    // per-wave h tiles, [edge][k]
  __shared__ int   dsts[8][32];

  const int tid = threadIdx.x;
  // Stage MLP weights in LDS once per block.
  for (int i = tid; i < 36 * 8; i += 256) kw1s[i] = kw1[i];
  if (tid < 8) kb1s[tid] = kb1[tid];
  for (int i = tid; i < 8 * 256; i += 256) kw2s[i] = kw2[i];
  kb2s[tid] = kb2[tid];
  __syncthreads();

  const int w    = tid >> 5;
  const int lane = tid & 31;
  const int e    = blockIdx.x * 256 + tid;   // E == grid*block exactly
  const int src  = ei[e];
  const int dst  = ei[EE + e];

  // Gather attr = [x0[src](16), grid0[src](2), x0[dst](16), grid0[dst](2)].
  // xT rows are contiguous 64B -> compiler merges to global_load_b128 x4.
  float attr[36];
#pragma unroll
  for (int c = 0; c < 16; ++c) attr[c] = xT[src * 16 + c];
  attr[16] = grid[2 * src];
  attr[17] = grid[2 * src + 1];
#pragma unroll
  for (int c = 0; c < 16; ++c) attr[18 + c] = xT[dst * 16 + c];
  attr[34] = grid[2 * dst];
  attr[35] = grid[2 * dst + 1];

  // h = gelu(attr @ W1 + b1), exact (erf) gelu.
#pragma unroll
  for (int k = 0; k < 8; ++k) {
    float acc = kb1s[k];
#pragma unroll
    for (int a = 0; a < 36; ++a) acc += attr[a] * kw1s[a * 8 + k];
    hs[w][lane * 8 + k] = 0.5f * acc * (1.0f + erff(acc * 0.70710678118654752440f));
  }
  // Transposed x tile: xs[c][edge] so the per-chunk broadcast for v=0..7 is
  // contiguous in LDS (2x ds_load_b128, values land in msg[v] order).
#pragma unroll
  for (int c = 0; c < 16; ++c) xs[w][c * 32 + lane] = attr[c];
  dsts[w][lane] = dst;
  atomicAdd(&cnt[dst], 1.0f);
  __syncthreads();

  // WMMA phase: two 16-edge subtiles per wave.
  const int m    = lane & 15;
  const int half = lane >> 4;
  for (int st = 0; st < 2; ++st) {
    const int eb = st * 16;
    // A operand (Htile 16x8, split K into two 16x16x4 steps), ISA A-layout:
    // lane L holds A[L%16][2*(L/16) + {0,1}].
    v2f A0, A1;
    A0.x = hs[w][(eb + m) * 8 + (2 * half + 0)];
    A0.y = hs[w][(eb + m) * 8 + (2 * half + 1)];
    A1.x = hs[w][(eb + m) * 8 + (4 + 2 * half + 0)];
    A1.y = hs[w][(eb + m) * 8 + (4 + 2 * half + 1)];

    v8f msg = {0, 0, 0, 0, 0, 0, 0, 0};
    for (int j = 0; j < 16; ++j) {   // output chunk j == input-channel c of W_e
      v2f B0, B1;                    // W2 columns [16j .. 16j+15], K-split
      B0.x = kw2s[(0 + 2 * half) * 256 + 16 * j + m];
      B0.y = kw2s[(1 + 2 * half) * 256 + 16 * j + m];
      B1.x = kw2s[(4 + 2 * half) * 256 + 16 * j + m];
      B1.y = kw2s[(5 + 2 * half) * 256 + 16 * j + m];
      const float cb = kb2s[16 * j + m];
      v8f c = {cb, cb, cb, cb, cb, cb, cb, cb};
      c = __builtin_amdgcn_wmma_f32_16x16x4_f32(false, A0, false, B0,
                                                (short)0, c, false, false);
      c = __builtin_amdgcn_wmma_f32_16x16x4_f32(false, A1, false, B1,
                                                (short)0, c, false, false);
      // Fused per-edge matvec: msg[e][o] += x[e][j] * W_e[j][o].
      // D layout: VGPR v / lane L -> (edge = v + 8*half, o = L%16).
      // xs is [c][edge]: addresses for v=0..7 are contiguous.
#pragma unroll
      for (int v = 0; v < 8; ++v) {
        const float xv = xs[w][j * 32 + eb + v + 8 * half];  // LDS broadcast
        msg[v] = fmaf(xv, c[v], msg[v]);
      }
    }
    // Scatter-add: 16 lanes write consecutive channels of one agg row.
#pragma unroll
    for (int v = 0; v < 8; ++v) {
      const int dv = dsts[w][eb + v + 8 * half];
      atomicAdd(&agg[dv * 16 + m], msg[v]);
    }
  }
}

// ---------------------------------------------------------------------------
// K2: out[b][o][n] = (b==0 ? agg[n]/max(cnt,1) : 0) + (x[b,n] @ root_w).
// Computed transposed: D' = root_w^T x X^T so stores are lane-contiguous in n.
// One 16-row tile per wave, 8 tiles per block.
// ---------------------------------------------------------------------------
__global__ __launch_bounds__(256) void out_kernel(
    const float* __restrict__ u,    // [B][16][N]
    const float* __restrict__ rw,   // [16][16]
    const float* __restrict__ agg,  // [N][16]
    const float* __restrict__ cnt,  // [N]
    float* __restrict__ out) {      // [B][16][N]
  const int tid  = threadIdx.x;
  const int w    = tid >> 5;
  const int lane = tid & 31;
  const int m    = lane & 15;
  const int half = lane >> 4;
  const int tile = blockIdx.x * 8 + w;
  const int rows = tile * 16;       // flat row base (b*N + n)
  const int b    = rows / NN;
  const int nb   = rows - b * NN;
  const float* ub = u + b * 16 * NN;

  v8f d = {0, 0, 0, 0, 0, 0, 0, 0};
#pragma unroll
  for (int kk = 0; kk < 4; ++kk) {
    const int c0 = 4 * kk + 2 * half;
    v2f A, Bv;
    A.x  = rw[(c0 + 0) * 16 + m];       // A'[o=m][c], i.e. root_w transposed
    A.y  = rw[(c0 + 1) * 16 + m];
    Bv.x = ub[(c0 + 0) * NN + nb + m];  // B'[c][row=m] = x[row][c]
    Bv.y = ub[(c0 + 1) * NN + nb + m];
    d = __builtin_amdgcn_wmma_f32_16x16x4_f32(false, A, false, Bv,
                                              (short)0, d, false, false);
  }

  if (b == 0) {  // only batch-0 rows receive aggregated messages
    const int r = rows + m;            // row index per lane (N-dim of D')
    const float inv = 1.0f / fmaxf(cnt[r], 1.0f);
#pragma unroll
    for (int v = 0; v < 8; ++v) {
      const int o = v + 8 * half;      // D' M-dim = output channel
      d[v] += agg[r * 16 + o] * inv;
    }
  }
#pragma unroll
  for (int v = 0; v < 8; ++v) {
    const int o = v + 8 * half;
    out[(b * 16 + o) * NN + nb + m] = d[v];  // contiguous in n across lanes
  }
}

// ---------------------------------------------------------------------------
extern "C" void kernel_launch(void* const* d_in, const int* in_sizes, int n_in,
                              void* d_out, int out_size, void* d_ws, size_t ws_size,
                              hipStream_t stream) {
  const float* u    = (const float*)d_in[0];
  const float* grid = (const float*)d_in[1];
  const int*   ei   = (const int*)d_in[2];
  const float* kw1  = (const float*)d_in[3];
  const float* kb1  = (const float*)d_in[4];
  const float* kw2  = (const float*)d_in[5];
  const float* kb2  = (const float*)d_in[6];
  const float* rw   = (const float*)d_in[7];
  float* out = (float*)d_out;

  float* agg = (float*)d_ws;        // [N][16]
  float* cnt = agg + NN * 16;       // [N]
  float* xT  = cnt + NN;            // [N][16] transposed batch-0 features

  zero_ws_kernel<<<256, 256, 0, stream>>>(agg, NN * 17);
  transpose_x0_kernel<<<NN / 256, 256, 0, stream>>>(u, xT);
  edge_kernel<<<EE / 256, 256, 0, stream>>>(xT, grid, ei, kw1, kb1, kw2, kb2,
                                            agg, cnt);
  out_kernel<<<(BB * NN / 16) / 8, 256, 0, stream>>>(u, rw, agg, cnt, out);
}